// DeConv2d_17136919511113
// MI455X (gfx1250) — compile-verified
//
#include <hip/hip_runtime.h>

typedef __attribute__((ext_vector_type(2))) float v2f;
typedef __attribute__((ext_vector_type(8))) float v8f;
typedef __attribute__((ext_vector_type(2))) unsigned u32x2;
typedef __attribute__((ext_vector_type(4))) unsigned u32x4;
typedef __attribute__((ext_vector_type(8))) unsigned u32x8;

#define OC  16
#define IC  16
#define HID 16
#define KK  4
#define TPW 4          // 16-pixel tiles per wave (one 64-wide image row)
#define PSTRIDE 18     // padded LDS row stride in floats (bank-conflict-free)

// LDS layout in floats
#define LDS_W2P 0      // [IC*HID][PSTRIDE]  4608  (W2 rows padded 16->18)
#define LDS_W3P 4608   // [IC*KK][PSTRIDE]   1152  (W3 rows padded 16->18)
#define LDS_W1  5760   // [IC][HID]           256
#define LDS_B1  6016   // [IC][HID]           256
#define LDS_B2  6272   // [IC][HID]           256
#define LDS_B3  6528   // [IC][KK]             64
#define LDS_TOT 6592   // 26368 bytes

#define USE_PERMLANE 1

// Issue one TDM descriptor copying nElems contiguous f32 from global to LDS.
// Optionally pad LDS destination by 2 DWORDs every 16 DWORDs (row 16 -> 18).
__device__ __forceinline__ void tdm_load_1d(unsigned lds_byte_off,
                                            const void* gsrc,
                                            unsigned nElems, bool pad16by2) {
    unsigned long long ga = (unsigned long long)(uintptr_t)gsrc;
    u32x4 g0;
    g0[0] = 1u;                                                // count=1, user D#
    g0[1] = lds_byte_off;                                      // lds_addr
    g0[2] = (unsigned)ga;                                      // global_addr[31:0]
    g0[3] = (unsigned)((ga >> 32) & 0x01FFFFFFu) | (2u << 30); // addr[56:32] | type=2
    u32x8 g1;
    g1[0] = (2u << 16)                                         // data_size=4B
          | (pad16by2 ? ((1u << 20) | (3u << 22) | (1u << 25)) : 0u);
            // pad_enable, pad_interval=3 (16 DW), pad_amount=1 (2 DW)
    g1[1] = (nElems & 0xFFFFu) << 16;     // tensor_dim0[15:0]
    g1[2] = (nElems >> 16) | (1u << 16);  // tensor_dim0[31:16], tensor_dim1=1
    g1[3] = (nElems & 0xFFFFu) << 16;     // tensor_dim1 hi=0, tile_dim0=nElems
    g1[4] = 1u;                           // tile_dim1=1, tile_dim2=0
    g1[5] = nElems;                       // tensor_dim0_stride[31:0]
    g1[6] = 0u;
    g1[7] = 0u;
    asm volatile("tensor_load_to_lds %0, %1" :: "s"(g0), "s"(g1) : "memory");
}

// Exchange upper half-wave of `a` with lower half-wave of `b`.
// After the call (lane view): a = {lanes0-15: old a, lanes16-31: old b[lane-16]}
//                             b = {lanes0-15: old a[lane+16], lanes16-31: old b}
__device__ __forceinline__ void half_swap(float& a, float& b, int hi) {
#if USE_PERMLANE
    u32x2 r = __builtin_amdgcn_permlane16_swap(
        __float_as_uint(a), __float_as_uint(b), false, false);
    a = __uint_as_float(r.x);
    b = __uint_as_float(r.y);
#else
    const float sa = __shfl_xor(a, 16, 32);
    const float sb = __shfl_xor(b, 16, 32);
    const float na = hi ? sb : a;
    const float nb = hi ? b : sa;
    a = na;
    b = nb;
#endif
}

// Block = 8 waves sharing one output channel o (weights staged in LDS by the
// Tensor Data Mover with padded rows). Each wave owns one 64-pixel image row
// = 4 WMMA tiles; weight fragments are registered once per i and reused x4.
__global__ __launch_bounds__(256) void deconv_mlp_wmma(
    const float* __restrict__ batches,  // [n, IC, iH, iW]
    const float* __restrict__ W1,       // [OC, IC, HID]
    const float* __restrict__ b1,       // [OC, IC, HID]
    const float* __restrict__ W2,       // [OC, IC, HID, HID]  (g, h)
    const float* __restrict__ b2,       // [OC, IC, HID]
    const float* __restrict__ W3,       // [OC, IC, KK, HID]   (k, h)
    const float* __restrict__ b3,       // [OC, IC, KK]
    float* __restrict__ out,            // [n, OC, 2*iH, 2*iW]
    int nBatch, int iH, int iW)
{
    __shared__ float smem[LDS_TOT];

    const int lane        = threadIdx.x & 31;
    const int waveInBlk   = threadIdx.x >> 5;
    const int wavesPerBlk = blockDim.x >> 5;

    const int o       = blockIdx.x & (OC - 1);
    const int rowBase = (blockIdx.x >> 4) * wavesPerBlk;

    // ---- Stage this o's weight slab into LDS via the Tensor Data Mover ----
    if (waveInBlk == 0) {
        const unsigned base = (unsigned)(uintptr_t)smem;   // LDS byte offset
        tdm_load_1d(base + LDS_W2P * 4u, W2 + (size_t)o * IC * HID * HID,
                    IC * HID * HID, true);
        tdm_load_1d(base + LDS_W3P * 4u, W3 + o * IC * KK * HID,
                    IC * KK * HID, true);
        tdm_load_1d(base + LDS_W1 * 4u, W1 + o * IC * HID, IC * HID, false);
        tdm_load_1d(base + LDS_B1 * 4u, b1 + o * IC * HID, IC * HID, false);
        tdm_load_1d(base + LDS_B2 * 4u, b2 + o * IC * HID, IC * HID, false);
        tdm_load_1d(base + LDS_B3 * 4u, b3 + o * IC * KK, IC * KK, false);
        __builtin_amdgcn_s_wait_tensorcnt(0);
    }
    __syncthreads();

    const int nRows = nBatch * iH;                    // one wave per image row
    const int row = rowBase + waveInBlk;
    if (row >= nRows) return;                         // wave-uniform guard

    const int ty = row % iH;
    const int nb = row / iH;

    const int col   = lane & 15;     // pixel within tile == WMMA N column == A/M row
    const int hi    = lane >> 4;     // half-wave select
    const int hsel  = hi * 2;        // K offset inside a 4-chunk (A/B VGPR layout)
    const int gbase = 8 * hi;        // C/D row base for this half-wave
    const int kc    = col & 3;       // clamped W3 row (masked, no EXEC change)
    const float kmask = (col < 4) ? 1.0f : 0.0f;

    const float* xrow = batches + ((size_t)nb * IC * iH + ty) * iW + col;
    const size_t HW = (size_t)iH * iW;

    v8f acc[TPW];
    #pragma unroll
    for (int t = 0; t < TPW; ++t) acc[t] = (v8f){};
    float b3s0 = 0.f, b3s1 = 0.f, b3s2 = 0.f, b3s3 = 0.f;

    for (int i = 0; i < IC; ++i) {
        // ---- hoist this i's weight fragments into registers (reused x4) ----
        v2f a2[4], a3[4], w1v[4], b1v[4];
        #pragma unroll
        for (int c = 0; c < 4; ++c) {
            const int hb = 4 * c + hsel;
            a2[c].x = smem[LDS_W2P + (i * HID + col) * PSTRIDE + hb];
            a2[c].y = smem[LDS_W2P + (i * HID + col) * PSTRIDE + hb + 1];
            a3[c].x = smem[LDS_W3P + (i * KK + kc) * PSTRIDE + hb]     * kmask;
            a3[c].y = smem[LDS_W3P + (i * KK + kc) * PSTRIDE + hb + 1] * kmask;
            w1v[c].x = smem[LDS_W1 + i * HID + hb];
            w1v[c].y = smem[LDS_W1 + i * HID + hb + 1];
            b1v[c].x = smem[LDS_B1 + i * HID + hb];
            b1v[c].y = smem[LDS_B1 + i * HID + hb + 1];
        }
        float b2v[8];
        #pragma unroll
        for (int r = 0; r < 8; ++r) b2v[r] = smem[LDS_B2 + i * HID + gbase + r];

        // x values for the 4 tiles of this row (coalesced b32 loads)
        float xv[TPW];
        #pragma unroll
        for (int t = 0; t < TPW; ++t) xv[t] = xrow[i * HW + t * 16];

        #pragma unroll
        for (int t = 0; t < TPW; ++t) {
            // ---- Layer 2: c2 = W2 x H1^T (4 chained 16x16x4 f32 WMMAs) ----
            v8f c2 = {};
            #pragma unroll
            for (int c = 0; c < 4; ++c) {
                v2f bm;
                bm.x = fmaxf(fmaf(xv[t], w1v[c].x, b1v[c].x), 0.0f);
                bm.y = fmaxf(fmaf(xv[t], w1v[c].y, b1v[c].y), 0.0f);
                c2 = __builtin_amdgcn_wmma_f32_16x16x4_f32(
                        false, a2[c], false, bm, (short)0, c2, false, false);
            }
            // bias + relu in C/D layout: VGPR r = row g = r + 8*hi
            #pragma unroll
            for (int r = 0; r < 8; ++r)
                c2[r] = fmaxf(c2[r] + b2v[r], 0.0f);

            // ---- transpose C/D layout -> B layout via half-wave swaps ----
            float p0 = c2[0], p2 = c2[2];  half_swap(p0, p2, hi);  // cc0/cc2 .x
            float p1 = c2[1], p3 = c2[3];  half_swap(p1, p3, hi);  // cc0/cc2 .y
            float q0 = c2[4], q2 = c2[6];  half_swap(q0, q2, hi);  // cc1/cc3 .x
            float q1 = c2[5], q3 = c2[7];  half_swap(q1, q3, hi);  // cc1/cc3 .y

            // ---- Layer 3: acc[t] += W3pad x H2^T (accumulates over i) ----
            v2f bm0; bm0.x = p0; bm0.y = p1;
            v2f bm1; bm1.x = q0; bm1.y = q1;
            v2f bm2; bm2.x = p2; bm2.y = p3;
            v2f bm3; bm3.x = q2; bm3.y = q3;
            acc[t] = __builtin_amdgcn_wmma_f32_16x16x4_f32(
                        false, a3[0], false, bm0, (short)0, acc[t], false, false);
            acc[t] = __builtin_amdgcn_wmma_f32_16x16x4_f32(
                        false, a3[1], false, bm1, (short)0, acc[t], false, false);
            acc[t] = __builtin_amdgcn_wmma_f32_16x16x4_f32(
                        false, a3[2], false, bm2, (short)0, acc[t], false, false);
            acc[t] = __builtin_amdgcn_wmma_f32_16x16x4_f32(
                        false, a3[3], false, bm3, (short)0, acc[t], false, false);
        }

        b3s0 += smem[LDS_B3 + i * KK + 0];
        b3s1 += smem[LDS_B3 + i * KK + 1];
        b3s2 += smem[LDS_B3 + i * KK + 2];
        b3s3 += smem[LDS_B3 + i * KK + 3];
    }

    // ---- store: rows k=0..3 at lanes 0-15; swap kh=1 pair to upper half ----
    const int oH = iH << 1;
    const int oW = iW << 1;
    const size_t rowOff =
        ((size_t)(nb * OC + o) * oH + (ty << 1) + hi) * oW + (col << 1);

    #pragma unroll
    for (int t = 0; t < TPW; ++t) {
        float s0 = acc[t][0] + b3s0;   // (kh=0, kw=0)
        float s1 = acc[t][1] + b3s1;   // (kh=0, kw=1)
        float s2 = acc[t][2] + b3s2;   // (kh=1, kw=0)
        float s3 = acc[t][3] + b3s3;   // (kh=1, kw=1)
        half_swap(s0, s2, hi);         // s0 = kw-pair for this half's output row
        half_swap(s1, s3, hi);
        v2f st; st.x = s0; st.y = s1;
        *(v2f*)(out + rowOff + 32 * t) = st;
    }
}

extern "C" void kernel_launch(void* const* d_in, const int* in_sizes, int n_in,
                              void* d_out, int out_size, void* d_ws, size_t ws_size,
                              hipStream_t stream) {
    const float* batches = (const float*)d_in[0];
    const float* W1 = (const float*)d_in[1];
    const float* b1 = (const float*)d_in[2];
    const float* W2 = (const float*)d_in[3];
    const float* b2 = (const float*)d_in[4];
    const float* W3 = (const float*)d_in[5];
    const float* b3 = (const float*)d_in[6];
    float* out = (float*)d_out;

    const int iH = 64, iW = 64;
    const int nBatch = in_sizes[0] / (IC * iH * iW);   // 8 for the reference
    const int nRows  = nBatch * iH;                    // one wave per image row
    const int wavesPerBlk = 256 / 32;
    const int rowGroups = (nRows + wavesPerBlk - 1) / wavesPerBlk;
    const int blocks = rowGroups * OC;                 // o = blockIdx & 15

    deconv_mlp_wmma<<<blocks, 256, 0, stream>>>(
        batches, W1, b1, W2, b2, W3, b3, out, nBatch, iH, iW);
}